// KalmanFilter_85194971283763
// MI455X (gfx1250) — compile-verified
//
#include <hip/hip_runtime.h>

typedef __attribute__((ext_vector_type(2))) float v2f;
typedef __attribute__((ext_vector_type(4))) float v4f;
typedef __attribute__((ext_vector_type(8))) float v8f;

#define BN 512
#define TN 200
#define SN 32
#define MN 4

// D = A(16x4) * B(4x16) + C, fp32 WMMA (CDNA5 V_WMMA_F32_16X16X4_F32)
__device__ __forceinline__ v8f wmma4(v2f a, v2f b, v8f c) {
  return __builtin_amdgcn_wmma_f32_16x16x4_f32(false, a, false, b, (short)0, c,
                                               false, false);
}

__global__ __launch_bounds__(32) void kalman_scan_kernel(
    const float* __restrict__ obs, const float* __restrict__ Fg,
    const float* __restrict__ Hg, const float* __restrict__ Qg,
    const float* __restrict__ Rg, const float* __restrict__ P0g,
    float* __restrict__ meansOut, float* __restrict__ covsOut) {
  __shared__ float sCov[SN][SN];   // current covariance (row major)
  __shared__ float sTmp[SN][SN];   // scratch (F*Pu, then Pp for symmetrize)
  __shared__ float sF[SN][SN];
  __shared__ float sH[MN][SN];
  __shared__ float sHP[MN][SN];
  __shared__ float sKTn[MN][SN];   // -K^T  (negated for D = (-K^T)*HP + P)
  __shared__ float sS[MN * MN];
  __shared__ float sResid[MN];
  __shared__ float sMean[SN];
  __shared__ float sMeanU[SN];

  const int b = blockIdx.x;
  const int lane = threadIdx.x;    // 0..31 (wave32)
  const int half = lane >> 4;      // 0 | 1
  const int l15 = lane & 15;

  const float* Fb = Fg + (size_t)b * SN * SN;
  const float* Hb = Hg + (size_t)b * MN * SN;
  const float* Qb = Qg + (size_t)b * SN * SN;
  const float* Rb = Rg + (size_t)b * MN * MN;
  const float* Pb = P0g + (size_t)b * SN * SN;
  const float* obsb = obs + (size_t)b * TN * MN;
  float* meanB = meansOut + (size_t)b * TN * SN;
  float* covB = covsOut + (size_t)b * TN * SN * SN;

  // ---- loop-invariant loads ----
#pragma unroll
  for (int r = 0; r < SN; ++r) {
    sF[r][lane] = Fb[r * SN + lane];
    sCov[r][lane] = Pb[r * SN + lane];
  }
#pragma unroll
  for (int m = 0; m < MN; ++m) sH[m][lane] = Hb[m * SN + lane];
  sMean[lane] = 0.0f;

  // F as A-fragments; identical data also serves as B-fragments of F^T
  // A[m][k]: lane m=l15, k = 2*half + vgpr
  v2f FA[2][8];
#pragma unroll
  for (int ti = 0; ti < 2; ++ti)
#pragma unroll
    for (int kc = 0; kc < 8; ++kc) {
      const float* row = Fb + (size_t)(ti * 16 + l15) * SN + kc * 4 + 2 * half;
      FA[ti][kc].x = row[0];
      FA[ti][kc].y = row[1];
    }
  // Q as C-fragments: vgpr v -> row v + 8*half, col l15
  v8f QC[2][2];
#pragma unroll
  for (int ti = 0; ti < 2; ++ti)
#pragma unroll
    for (int tj = 0; tj < 2; ++tj)
#pragma unroll
      for (int v = 0; v < 8; ++v)
        QC[ti][tj][v] = Qb[(size_t)(ti * 16 + half * 8 + v) * SN + tj * 16 + l15];
  __syncthreads();

  for (int t = 0; t < TN; ++t) {
    // ---- record 1-step-ahead belief (pre-update) ----
    __builtin_nontemporal_store(sMean[lane], meanB + t * SN + lane);
    float* crow = covB + (size_t)t * SN * SN + lane * SN;  // lane owns row `lane`
#pragma unroll
    for (int j4 = 0; j4 < 8; ++j4) {
      v4f vv = {sCov[lane][4 * j4 + 0], sCov[lane][4 * j4 + 1],
                sCov[lane][4 * j4 + 2], sCov[lane][4 * j4 + 3]};
      __builtin_nontemporal_store(vv, (v4f*)crow + j4);
    }

    // ---- HP = H * P   (lane owns column s=lane) ----
    float hp[MN];
#pragma unroll
    for (int m = 0; m < MN; ++m) {
      float a = 0.f;
#pragma unroll
      for (int k = 0; k < SN; ++k) a += sH[m][k] * sCov[k][lane];
      hp[m] = a;
      sHP[m][lane] = a;
    }
    __syncthreads();

    // ---- S = HP*H^T + R (lanes 0..15), resid = obs - H*mean (lanes 0..3) ----
    if (lane < 16) {
      const int mi = lane >> 2, ni = lane & 3;
      float a = Rb[mi * MN + ni];
#pragma unroll
      for (int s = 0; s < SN; ++s) a += sHP[mi][s] * sH[ni][s];
      sS[lane] = a;
    }
    if (lane < MN) {
      float a = obsb[t * MN + lane];
#pragma unroll
      for (int s = 0; s < SN; ++s) a -= sH[lane][s] * sMean[s];
      sResid[lane] = a;
    }
    __syncthreads();

    // ---- per-lane 4x4 LU solve: K^T[:,lane] = S^{-1} * HP[:,lane] ----
    float A4[4][4], rhs[4];
#pragma unroll
    for (int i = 0; i < 4; ++i) {
      rhs[i] = hp[i];
#pragma unroll
      for (int j = 0; j < 4; ++j) A4[i][j] = sS[i * 4 + j];
    }
#pragma unroll
    for (int kk = 0; kk < 4; ++kk) {
      const float inv = 1.f / A4[kk][kk];
#pragma unroll
      for (int i = 0; i < 4; ++i) {
        if (i > kk) {
          const float f = A4[i][kk] * inv;
#pragma unroll
          for (int j = 0; j < 4; ++j)
            if (j > kk) A4[i][j] -= f * A4[kk][j];
          rhs[i] -= f * rhs[kk];
        }
      }
    }
    float kt[4];
#pragma unroll
    for (int kk = 3; kk >= 0; --kk) {
      float a = rhs[kk];
#pragma unroll
      for (int j = 0; j < 4; ++j)
        if (j > kk) a -= A4[kk][j] * kt[j];
      kt[kk] = a / A4[kk][kk];
    }
#pragma unroll
    for (int m = 0; m < MN; ++m) sKTn[m][lane] = -kt[m];

    // ---- mean update ----
    float mu = sMean[lane];
#pragma unroll
    for (int m = 0; m < MN; ++m) mu += kt[m] * sResid[m];
    sMeanU[lane] = mu;
    __syncthreads();

    // ---- Pu = P + (-K^T)^T(32x4) * HP(4x32) : one K=4 WMMA per 16x16 tile --
    v8f cu[2][2];
#pragma unroll
    for (int ti = 0; ti < 2; ++ti)
#pragma unroll
      for (int tj = 0; tj < 2; ++tj)
#pragma unroll
        for (int v = 0; v < 8; ++v)
          cu[ti][tj][v] = sCov[ti * 16 + half * 8 + v][tj * 16 + l15];
    v2f aKT[2], bHP[2];
#pragma unroll
    for (int ti = 0; ti < 2; ++ti) {
      aKT[ti].x = sKTn[2 * half + 0][ti * 16 + l15];
      aKT[ti].y = sKTn[2 * half + 1][ti * 16 + l15];
      bHP[ti].x = sHP[2 * half + 0][ti * 16 + l15];
      bHP[ti].y = sHP[2 * half + 1][ti * 16 + l15];
    }
#pragma unroll
    for (int ti = 0; ti < 2; ++ti)
#pragma unroll
      for (int tj = 0; tj < 2; ++tj)
        cu[ti][tj] = wmma4(aKT[ti], bHP[tj], cu[ti][tj]);
    __syncthreads();
#pragma unroll
    for (int ti = 0; ti < 2; ++ti)
#pragma unroll
      for (int tj = 0; tj < 2; ++tj)
#pragma unroll
        for (int v = 0; v < 8; ++v)
          sCov[ti * 16 + half * 8 + v][tj * 16 + l15] = cu[ti][tj][v];
    __syncthreads();

    // ---- tmp = F * Pu : 2x2 tiles x 8 K-chunks of fp32 WMMA ----
#pragma unroll
    for (int ti = 0; ti < 2; ++ti)
#pragma unroll
      for (int tj = 0; tj < 2; ++tj) {
        v8f acc = {};
#pragma unroll
        for (int kc = 0; kc < 8; ++kc) {
          v2f bb;
          bb.x = sCov[kc * 4 + 2 * half + 0][tj * 16 + l15];
          bb.y = sCov[kc * 4 + 2 * half + 1][tj * 16 + l15];
          acc = wmma4(FA[ti][kc], bb, acc);
        }
#pragma unroll
        for (int v = 0; v < 8; ++v)
          sTmp[ti * 16 + half * 8 + v][tj * 16 + l15] = acc[v];
      }
    __syncthreads();

    // ---- Pp = tmp * F^T + Q  (B-fragment of F^T == A-fragment of F) ----
    v8f cp[2][2];
#pragma unroll
    for (int ti = 0; ti < 2; ++ti)
#pragma unroll
      for (int tj = 0; tj < 2; ++tj) {
        v8f acc = QC[ti][tj];
#pragma unroll
        for (int kc = 0; kc < 8; ++kc) {
          v2f aa;
          aa.x = sTmp[ti * 16 + l15][kc * 4 + 2 * half + 0];
          aa.y = sTmp[ti * 16 + l15][kc * 4 + 2 * half + 1];
          acc = wmma4(aa, FA[tj][kc], acc);
        }
        cp[ti][tj] = acc;
      }
    __syncthreads();
#pragma unroll
    for (int ti = 0; ti < 2; ++ti)
#pragma unroll
      for (int tj = 0; tj < 2; ++tj)
#pragma unroll
        for (int v = 0; v < 8; ++v)
          sTmp[ti * 16 + half * 8 + v][tj * 16 + l15] = cp[ti][tj][v];
    __syncthreads();

    // ---- symmetrize Pp -> sCov ; mean_p = F * mean_u ----
    float mp = 0.f;
#pragma unroll
    for (int j = 0; j < SN; ++j) {
      sCov[lane][j] = 0.5f * (sTmp[lane][j] + sTmp[j][lane]);
      mp += sF[lane][j] * sMeanU[j];
    }
    __syncthreads();
    sMean[lane] = mp;
    __syncthreads();
  }
}

// ---- broadcast outputs: H_stack [B,T,M,S], R_stack [B,T,M,M] ----
__global__ void stack_H_kernel(const v4f* __restrict__ H4, v4f* __restrict__ out4,
                               int n4) {
  int i = blockIdx.x * blockDim.x + threadIdx.x;
  if (i >= n4) return;
  const int chunk = MN * SN / 4;  // 32 v4f per (b,t)
  int b = i / (TN * chunk);
  int j = i % chunk;
  __builtin_nontemporal_store(H4[b * chunk + j], out4 + i);
}

__global__ void stack_R_kernel(const v4f* __restrict__ R4, v4f* __restrict__ out4,
                               int n4) {
  int i = blockIdx.x * blockDim.x + threadIdx.x;
  if (i >= n4) return;
  const int chunk = MN * MN / 4;  // 4 v4f per (b,t)
  int b = i / (TN * chunk);
  int j = i % chunk;
  __builtin_nontemporal_store(R4[b * chunk + j], out4 + i);
}

extern "C" void kernel_launch(void* const* d_in, const int* in_sizes, int n_in,
                              void* d_out, int out_size, void* d_ws, size_t ws_size,
                              hipStream_t stream) {
  (void)in_sizes; (void)n_in; (void)out_size; (void)d_ws; (void)ws_size;
  const float* obs = (const float*)d_in[0];  // [B,T,M]
  const float* F = (const float*)d_in[1];    // [B,S,S]
  const float* H = (const float*)d_in[2];    // [B,M,S]
  const float* Q = (const float*)d_in[3];    // [B,S,S]
  const float* R = (const float*)d_in[4];    // [B,M,M]
  const float* P0 = (const float*)d_in[5];   // [B,S,S]

  float* means = (float*)d_out;                                   // [B,T,S]
  float* covs = means + (size_t)BN * TN * SN;                     // [B,T,S,S]
  float* Rout = covs + (size_t)BN * TN * SN * SN;                 // [B,T,M,M]
  float* Hout = Rout + (size_t)BN * TN * MN * MN;                 // [B,T,M,S]

  kalman_scan_kernel<<<BN, 32, 0, stream>>>(obs, F, H, Q, R, P0, means, covs);

  int n4h = BN * TN * (MN * SN / 4);
  stack_H_kernel<<<(n4h + 255) / 256, 256, 0, stream>>>((const v4f*)H,
                                                        (v4f*)Hout, n4h);
  int n4r = BN * TN * (MN * MN / 4);
  stack_R_kernel<<<(n4r + 255) / 256, 256, 0, stream>>>((const v4f*)R,
                                                        (v4f*)Rout, n4r);
}